// ContrastiveGCN_40063454937414
// MI455X (gfx1250) — compile-verified
//
#include <hip/hip_runtime.h>
#include <hip/hip_bf16.h>

typedef __attribute__((ext_vector_type(2))) float v2f;
typedef __attribute__((ext_vector_type(8))) float v8f;

#define NN   50000
#define NE   800000
#define INF_ 192
#define HID_ 128
#define OUT_ 64
#define FCO_ 192
#define MT   5        // M-tiles per wave (3125 M-tiles = 5 * 625)

// ---------- helpers: monotone float<->uint encoding for atomic max ----------
__device__ __forceinline__ unsigned fenc(float f) {
  unsigned u = __float_as_uint(f);
  return (u & 0x80000000u) ? ~u : (u | 0x80000000u);
}
__device__ __forceinline__ float fdec(unsigned u) {
  return __uint_as_float((u & 0x80000000u) ? (u & 0x7FFFFFFFu) : ~u);
}

// ---------- fills ----------
__global__ void fill_f32(float* __restrict__ p, long long n) {
  long long i = blockIdx.x * (long long)blockDim.x + threadIdx.x;
  long long stride = (long long)gridDim.x * blockDim.x;
  for (; i < n; i += stride) p[i] = 0.0f;
}
__global__ void fill_u32(unsigned* __restrict__ p, long long n, unsigned v) {
  long long i = blockIdx.x * (long long)blockDim.x + threadIdx.x;
  long long stride = (long long)gridDim.x * blockDim.x;
  for (; i < n; i += stride) p[i] = v;
}

// ---------- WMMA FP32 GEMM: C[M,Nc] = A[M,K] * B[K,Nc] (+bias) ----------
// Each wave computes MT stacked 16x16 output tiles (80 rows x 16 cols) so one
// B fragment feeds MT independent back-to-back V_WMMA_F32_16X16X4_F32 ops.
__global__ void wmma_gemm_f32(const float* __restrict__ A, const float* __restrict__ B,
                              const float* __restrict__ bias, float* __restrict__ C,
                              int M, int K, int Nc) {
  const int wavesPerBlock = blockDim.x >> 5;
  const int group = blockIdx.x * wavesPerBlock + (threadIdx.x >> 5);
  const int ntn = Nc >> 4;
  const int ngm = M / (16 * MT);            // M-tile groups (exact: 3125/5 = 625)
  if (group >= ngm * ntn) return;           // wave-uniform exit (EXEC stays all-1)
  const int gm = group / ntn;
  const int tn = group - gm * ntn;
  const int lane = threadIdx.x & 31;
  const int half = lane >> 4;               // 0: lanes 0-15, 1: lanes 16-31
  const int l16  = lane & 15;
  const int rowBase = gm * (16 * MT) + l16;
  const int bcol = (tn << 4) + l16;

  v8f acc[MT];
  #pragma unroll
  for (int t = 0; t < MT; ++t) acc[t] = (v8f){};

  // A frag: lanes 0-15 hold K+0/K+1, lanes 16-31 hold K+2/K+3
  const float* ap = A + (long long)rowBase * K + (half << 1);
  const long long tileStride = (long long)16 * K;
  // B frag: VGPR0 = K(+0 lo / +2 hi), VGPR1 = K(+1 lo / +3 hi)
  const float* bp = B + (long long)(half << 1) * Nc + bcol;

  #pragma unroll 2
  for (int k = 0; k < K; k += 4) {
    v2f b;
    b.x = bp[(long long)k * Nc];
    b.y = bp[(long long)(k + 1) * Nc];
    v2f a[MT];
    #pragma unroll
    for (int t = 0; t < MT; ++t)
      a[t] = *(const v2f*)(ap + t * tileStride + k);
    #pragma unroll
    for (int t = 0; t < MT; ++t)
      acc[t] = __builtin_amdgcn_wmma_f32_16x16x4_f32(false, a[t], false, b, (short)0,
                                                     acc[t], false, false);
  }

  const float bv = bias ? bias[bcol] : 0.0f;
  #pragma unroll
  for (int t = 0; t < MT; ++t) {
    // C/D layout: VGPR v -> row M = v (lanes 0-15) / v+8 (lanes 16-31)
    float* cp = C + (long long)(gm * (16 * MT) + t * 16 + (half << 3)) * Nc + bcol;
    #pragma unroll
    for (int v = 0; v < 8; ++v)
      cp[(long long)v * Nc] = acc[t][v] + bv;
  }
}

// ---------- degree / norm ----------
__global__ void deg_kernel(const long long* __restrict__ ei, float* __restrict__ deg, int E) {
  int id = blockIdx.x * blockDim.x + threadIdx.x;
  if (id >= E) return;
  atomicAdd(&deg[(int)ei[E + id]], 1.0f);
}
__global__ void dinv_kernel(float* __restrict__ d, int N) {
  int i = blockIdx.x * blockDim.x + threadIdx.x;
  if (i >= N) return;
  d[i] = rsqrtf(d[i] + 1.0f);               // +1 self-loop, deg >= 1
}

// ---------- GCN scatter: out[dst] += h[src] * dinv[src]*dinv[dst] ----------
// grid.x = E+N edges (self-loops appended), blockDim.x = F
__global__ void scatter_gcn(const float* __restrict__ h, const long long* __restrict__ ei,
                            const float* __restrict__ dinv, float* __restrict__ out,
                            int E, int N, int F) {
  const long long id = blockIdx.x;
  int s, d;
  if (id < E) { s = (int)ei[id]; d = (int)ei[E + id]; }
  else        { s = d = (int)(id - E); }
  const float nrm = dinv[s] * dinv[d];
  const int f = threadIdx.x;
  atomicAdd(&out[(long long)d * F + f], h[(long long)s * F + f] * nrm);
}

// ---------- bias (+optional relu) ----------
__global__ void bias_act(float* __restrict__ h, const float* __restrict__ b,
                         long long total, int F, int doRelu) {
  long long i = blockIdx.x * (long long)blockDim.x + threadIdx.x;
  if (i >= total) return;
  float v = h[i] + b[(int)(i % F)];
  h[i] = doRelu ? fmaxf(v, 0.0f) : v;
}

// ---------- GAT attention ----------
__global__ void attn_logits(const float* __restrict__ hg, const float* __restrict__ a_src,
                            const float* __restrict__ a_dst, float* __restrict__ al,
                            float* __restrict__ ar, int N) {
  int i = blockIdx.x * blockDim.x + threadIdx.x;
  if (i >= N) return;
  const float* row = hg + (long long)i * HID_;
  float s = 0.0f, t = 0.0f;
  #pragma unroll 8
  for (int f = 0; f < HID_; ++f) { float v = row[f]; s += v * a_src[f]; t += v * a_dst[f]; }
  al[i] = s; ar[i] = t;
}

__device__ __forceinline__ void edge_sd(const long long* ei, long long id, int E, int& s, int& d) {
  if (id < E) { s = (int)ei[id]; d = (int)ei[E + id]; }
  else        { s = d = (int)(id - E); }
}
__device__ __forceinline__ float leaky02(float x) { return x > 0.0f ? x : 0.2f * x; }

__global__ void edge_max_kernel(const long long* __restrict__ ei, const float* __restrict__ al,
                                const float* __restrict__ ar, unsigned* __restrict__ emaxb,
                                int E, int N) {
  long long id = blockIdx.x * (long long)blockDim.x + threadIdx.x;
  if (id >= (long long)E + N) return;
  int s, d; edge_sd(ei, id, E, s, d);
  atomicMax(&emaxb[d], fenc(leaky02(al[s] + ar[d])));
}

__global__ void edge_exp_kernel(const long long* __restrict__ ei, const float* __restrict__ al,
                                const float* __restrict__ ar, const unsigned* __restrict__ emaxb,
                                float* __restrict__ denom, float* __restrict__ exbuf,
                                int E, int N) {
  long long id = blockIdx.x * (long long)blockDim.x + threadIdx.x;
  if (id >= (long long)E + N) return;
  int s, d; edge_sd(ei, id, E, s, d);
  float e  = leaky02(al[s] + ar[d]);
  float ex = expf(e - fdec(emaxb[d]));
  exbuf[id] = ex;
  atomicAdd(&denom[d], ex);
}

// grid.x = E+N, blockDim.x = HID_
__global__ void gat_agg(const long long* __restrict__ ei, const float* __restrict__ hg,
                        const float* __restrict__ exbuf, const float* __restrict__ denom,
                        float* __restrict__ out, int E, int N) {
  const long long id = blockIdx.x;
  int s, d; edge_sd(ei, id, E, s, d);
  const float alpha = exbuf[id] / denom[d];
  const int f = threadIdx.x;
  atomicAdd(&out[(long long)d * HID_ + f], hg[(long long)s * HID_ + f] * alpha);
}

extern "C" void kernel_launch(void* const* d_in, const int* in_sizes, int n_in,
                              void* d_out, int out_size, void* d_ws, size_t ws_size,
                              hipStream_t stream) {
  const float*      x     = (const float*)d_in[0];
  const long long*  ei    = (const long long*)d_in[1];   // int64 [2, E]
  const float*      W1    = (const float*)d_in[2];
  const float*      b1    = (const float*)d_in[3];
  const float*      Wg    = (const float*)d_in[4];
  const float*      a_src = (const float*)d_in[5];
  const float*      a_dst = (const float*)d_in[6];
  const float*      bg    = (const float*)d_in[7];
  const float*      W2    = (const float*)d_in[8];
  const float*      b2    = (const float*)d_in[9];
  const float*      Wf    = (const float*)d_in[10];
  const float*      bf    = (const float*)d_in[11];
  float*            out   = (float*)d_out;

  const int N = NN, E = NE;
  const long long EN = (long long)E + N;
  const int ngm = N / (16 * MT);            // 625 M-tile groups

  // workspace carve-out (256B aligned regions)
  char* ws = (char*)d_ws; size_t off = 0;
  auto carve = [&](size_t bytes) { void* p = ws + off; off += (bytes + 255) & ~(size_t)255; return p; };
  float*    dinv  = (float*)carve((size_t)N * sizeof(float));
  float*    bufA  = (float*)carve((size_t)N * HID_ * sizeof(float));   // h1 -> hg -> h2pre/h2
  float*    bufB  = (float*)carve((size_t)N * HID_ * sizeof(float));   // h -> hagg (relu'd)
  float*    al    = (float*)carve((size_t)N * sizeof(float));
  float*    ar    = (float*)carve((size_t)N * sizeof(float));
  unsigned* emaxb = (unsigned*)carve((size_t)N * sizeof(unsigned));
  float*    denom = (float*)carve((size_t)N * sizeof(float));
  float*    exbuf = (float*)carve((size_t)EN * sizeof(float));
  float*    h2a   = bufA;                                // N x 64
  float*    h2b   = bufA + (size_t)N * OUT_;             // N x 64

  // ---- degrees / symmetric norm ----
  fill_f32<<<1024, 256, 0, stream>>>(dinv, N);
  deg_kernel<<<(E + 255) / 256, 256, 0, stream>>>(ei, dinv, E);
  dinv_kernel<<<(N + 255) / 256, 256, 0, stream>>>(dinv, N);

  // ---- GCN1: h1 = x @ W1 ; scatter ; + b1 ----
  {
    int groups = ngm * (HID_ / 16);
    wmma_gemm_f32<<<(groups + 3) / 4, 128, 0, stream>>>(x, W1, nullptr, bufA, N, INF_, HID_);
  }
  fill_f32<<<4096, 256, 0, stream>>>(bufB, (long long)N * HID_);
  scatter_gcn<<<(unsigned)EN, HID_, 0, stream>>>(bufA, ei, dinv, bufB, E, N, HID_);
  bias_act<<<(unsigned)(((long long)N * HID_ + 255) / 256), 256, 0, stream>>>(
      bufB, b1, (long long)N * HID_, HID_, 0);

  // ---- GAT: hg = h @ Wg ; softmax over dst ; aggregate ; + bg ; relu ----
  {
    int groups = ngm * (HID_ / 16);
    wmma_gemm_f32<<<(groups + 3) / 4, 128, 0, stream>>>(bufB, Wg, nullptr, bufA, N, HID_, HID_);
  }
  attn_logits<<<(N + 255) / 256, 256, 0, stream>>>(bufA, a_src, a_dst, al, ar, N);
  fill_u32<<<1024, 256, 0, stream>>>(emaxb, N, 0x007FFFFFu);   // fenc(-inf)
  edge_max_kernel<<<(unsigned)((EN + 255) / 256), 256, 0, stream>>>(ei, al, ar, emaxb, E, N);
  fill_f32<<<1024, 256, 0, stream>>>(denom, N);
  edge_exp_kernel<<<(unsigned)((EN + 255) / 256), 256, 0, stream>>>(ei, al, ar, emaxb, denom, exbuf, E, N);
  fill_f32<<<4096, 256, 0, stream>>>(bufB, (long long)N * HID_);
  gat_agg<<<(unsigned)EN, HID_, 0, stream>>>(ei, bufA, exbuf, denom, bufB, E, N);
  bias_act<<<(unsigned)(((long long)N * HID_ + 255) / 256), 256, 0, stream>>>(
      bufB, bg, (long long)N * HID_, HID_, 1);

  // ---- GCN2: h2pre = relu(h) @ W2 ; scatter ; + b2 ----
  {
    int groups = ngm * (OUT_ / 16);
    wmma_gemm_f32<<<(groups + 3) / 4, 128, 0, stream>>>(bufB, W2, nullptr, h2a, N, HID_, OUT_);
  }
  fill_f32<<<4096, 256, 0, stream>>>(h2b, (long long)N * OUT_);
  scatter_gcn<<<(unsigned)EN, OUT_, 0, stream>>>(h2a, ei, dinv, h2b, E, N, OUT_);
  bias_act<<<(unsigned)(((long long)N * OUT_ + 255) / 256), 256, 0, stream>>>(
      h2b, b2, (long long)N * OUT_, OUT_, 0);

  // ---- FC: out = h2 @ Wf + bf ----
  {
    int groups = ngm * (FCO_ / 16);
    wmma_gemm_f32<<<(groups + 3) / 4, 128, 0, stream>>>(h2b, Wf, bf, out, N, OUT_, FCO_);
  }

  (void)in_sizes; (void)n_in; (void)out_size; (void)ws_size;
}